// ESBN_77352361001553
// MI455X (gfx1250) — compile-verified
//
#include <hip/hip_runtime.h>
#include <math.h>

typedef __attribute__((ext_vector_type(2))) float v2f;
typedef __attribute__((ext_vector_type(8))) float v8f;

#define HIDDEN 512
#define GATES  2048
#define TSTEPS 16
#define BATCH  1024

// One workgroup (512 threads = 16 wave32) runs the whole batch-degenerate
// LSTM recurrence. gates(2048) = Whh(2048x512) * h(512) via f32 WMMA 16x16x4,
// with h replicated across the 16 N-columns of B.
__launch_bounds__(512)
__global__ void esbn_recurrence(const float* __restrict__ Whh,
                                const float* __restrict__ bih,
                                const float* __restrict__ bhh,
                                const float* __restrict__ Wo,
                                const float* __restrict__ bo,
                                float* __restrict__ outSmall) {
  __shared__ float h_lds[HIDDEN];
  __shared__ float c_lds[HIDDEN];
  __shared__ float cg_lds[GATES];   // bih + bhh (const_gates, batch-independent)
  __shared__ float g_lds[GATES];    // raw Whh*h per step

  const int tid  = threadIdx.x;
  const int wave = tid >> 5;
  const int lane = tid & 31;

  for (int j = tid; j < GATES; j += 512) cg_lds[j] = bih[j] + bhh[j];
  h_lds[tid] = 0.f;
  c_lds[tid] = 0.f;
  __syncthreads();

  // f32 16x16x4 A-matrix layout: lanes 0-15 -> K=0,1 ; lanes 16-31 -> K=2,3
  const int koff = (lane < 16) ? 0 : 2;
  const int rowl = lane & 15;

  for (int t = 0; t < TSTEPS; ++t) {
    // ---- gates = Whh * h (each wave owns 8 row-tiles of 16) ----
    for (int tt = 0; tt < 8; ++tt) {
      const int tile16 = (wave * 8 + tt) * 16;
      const float* wrow = Whh + (size_t)(tile16 + rowl) * HIDDEN;
      v8f c = {};
      for (int k0 = 0; k0 < HIDDEN; k0 += 4) {
        v2f a, b;
        a.x = wrow[k0 + koff];
        a.y = wrow[k0 + koff + 1];
        b.x = h_lds[k0 + koff];        // replicated across N columns
        b.y = h_lds[k0 + koff + 1];
        c = __builtin_amdgcn_wmma_f32_16x16x4_f32(false, a, false, b,
                                                  (short)0, c, false, false);
      }
      // C/D layout: lanes 0-15 hold M=VGPR, lanes 16-31 hold M=VGPR+8.
      // All N columns identical -> lane 0 / lane 16 carry the 16 row values.
      if (lane == 0) {
#pragma unroll
        for (int v = 0; v < 8; ++v) g_lds[tile16 + v] = c[v];
      }
      if (lane == 16) {
#pragma unroll
        for (int v = 0; v < 8; ++v) g_lds[tile16 + 8 + v] = c[v];
      }
    }
    __syncthreads();

    // ---- elementwise LSTM cell (torch gate order i,f,g,o) ----
    {
      const int j = tid;
      float ig = g_lds[j]              + cg_lds[j];
      float fg = g_lds[HIDDEN + j]     + cg_lds[HIDDEN + j];
      float gg = g_lds[2 * HIDDEN + j] + cg_lds[2 * HIDDEN + j];
      float og = g_lds[3 * HIDDEN + j] + cg_lds[3 * HIDDEN + j];
      float si = 1.f / (1.f + __expf(-ig));
      float sf = 1.f / (1.f + __expf(-fg));
      float so = 1.f / (1.f + __expf(-og));
      float cn = sf * c_lds[j] + si * tanhf(gg);
      float hn = so * tanhf(cn);
      c_lds[j] = cn;
      h_lds[j] = hn;
    }
    __syncthreads();

    // ---- out[t] = h @ Wo.T + bo (4 values; tiny, do serially) ----
    if (tid < 4) {
      float s = bo[tid];
      const float* w = Wo + tid * HIDDEN;
      for (int k = 0; k < HIDDEN; ++k) s += h_lds[k] * w[k];
      outSmall[t * 4 + tid] = s;
    }
    __syncthreads();
  }
}

// Broadcast the 16x4 result across B=1024: out[t][b][0:4] = outSmall[t][0:4]
__global__ void esbn_broadcast(const float4* __restrict__ outSmall4,
                               float4* __restrict__ out4) {
  int idx = blockIdx.x * blockDim.x + threadIdx.x;  // idx = t*1024 + b
  if (idx < TSTEPS * BATCH) {
    out4[idx] = outSmall4[idx >> 10];
  }
}

extern "C" void kernel_launch(void* const* d_in, const int* in_sizes, int n_in,
                              void* d_out, int out_size, void* d_ws, size_t ws_size,
                              hipStream_t stream) {
  // setup_inputs order:
  // 0:images 1:W1 2:b1 3:W2 4:b2 5:W3 6:b3 7:We 8:be
  // 9:Wih 10:Whh 11:bih 12:bhh 13:Wg 14:bg 15:Wk 16:bk 17:Wo 18:bo 19:Wc 20:bc
  const float* Whh = (const float*)d_in[10];
  const float* bih = (const float*)d_in[11];
  const float* bhh = (const float*)d_in[12];
  const float* Wo  = (const float*)d_in[17];
  const float* bo  = (const float*)d_in[18];

  float* outSmall = (float*)d_ws;  // 64 floats

  esbn_recurrence<<<1, 512, 0, stream>>>(Whh, bih, bhh, Wo, bo, outSmall);
  esbn_broadcast<<<(TSTEPS * BATCH + 255) / 256, 256, 0, stream>>>(
      (const float4*)d_ws, (float4*)d_out);
}